// E44DiagonalWCell_47656957116867
// MI455X (gfx1250) — compile-verified
//
#include <hip/hip_runtime.h>
#include <cstdint>

typedef float f32x4 __attribute__((ext_vector_type(4)));

// ---- problem constants (match reference) ----
constexpr int T   = 4096;
constexpr int B   = 8;
constexpr int D   = 1024;
constexpr int BD  = B * D;          // 8192 channels
constexpr int CCH = 32;             // number of scan chunks
constexpr int L   = T / CCH;        // 128 timesteps per chunk
constexpr int G4  = BD / 4;         // 2048 float4 channel groups

// ---- phase-3 tiling ----
constexpr int CHB   = 1024;         // channels per block (4 per thread)
constexpr int TT    = 8;            // timesteps per LDS tile
constexpr int NTILE = L / TT;       // 16 tiles per chunk
// one LDS buffer = TT*CHB*4 = 32 KB; double buffered = 64 KB

__device__ __forceinline__ float sigf(float v) {          // precise (setup only)
    return 1.0f / (1.0f + __expf(-v));
}
__device__ __forceinline__ float fast_sig(float v) {      // 1-ulp rcp (hot path)
    return __builtin_amdgcn_rcpf(1.0f + __expf(-v));
}

// ---------------------------------------------------------------------------
// Phase 1: per (chunk, float4-channel-group) local scan with zero init.
// Regular-temporal x loads intentionally warm the 192 MB L2 (x = 128 MiB).
// ---------------------------------------------------------------------------
__global__ void __launch_bounds__(256) lru_phase1(
    const float* __restrict__ x, const float* __restrict__ logd,
    const float* __restrict__ bias, f32x4* __restrict__ sLast) {
    const int g     = blockIdx.x * 256 + threadIdx.x;   // 0..G4-1
    const int chunk = blockIdx.y;                       // 0..CCH-1
    const int dg    = g & (D / 4 - 1);                  // float4 group within D

    const f32x4 ld = ((const f32x4*)logd)[dg];
    const f32x4 bb = ((const f32x4*)bias)[dg];
    f32x4 dv;
#pragma unroll
    for (int j = 0; j < 4; ++j) dv[j] = sigf(ld[j]);

    f32x4 s = (f32x4)(0.f);
    const f32x4* xr = (const f32x4*)x;
    size_t idx = (size_t)(chunk * L) * (size_t)G4 + (size_t)g;
#pragma unroll 4
    for (int t = 0; t < L; ++t) {
        f32x4 xv = xr[idx];
        idx += (size_t)G4;
        s = dv * (xv + s) + bb;
    }
    sLast[(size_t)chunk * G4 + g] = s;
}

// ---------------------------------------------------------------------------
// Phase 2: sequential combine across chunks per channel group.
// carry[k][g] = h at entry of chunk k ; H_k = d^L * H_{k-1} + sLast_k.
// Also writes h[0,:,:] = h0.
// ---------------------------------------------------------------------------
__global__ void __launch_bounds__(256) lru_phase2(
    const float* __restrict__ h0, const float* __restrict__ logd,
    const f32x4* __restrict__ sLast, f32x4* __restrict__ carry,
    float* __restrict__ hout) {
    const int g  = blockIdx.x * 256 + threadIdx.x;      // 0..G4-1
    const int dg = g & (D / 4 - 1);

    const f32x4 ld = ((const f32x4*)logd)[dg];
    f32x4 dL;
#pragma unroll
    for (int j = 0; j < 4; ++j) dL[j] = sigf(ld[j]);
#pragma unroll
    for (int i = 0; i < 7; ++i) dL *= dL;               // d^(2^7) = d^L, L=128

    f32x4 H = ((const f32x4*)h0)[g];
    ((f32x4*)hout)[g] = H;                              // h[0] row
    for (int k = 0; k < CCH; ++k) {
        carry[(size_t)k * G4 + g] = H;
        H = dL * H + sLast[(size_t)k * G4 + g];
    }
}

// ---------------------------------------------------------------------------
// Phase 3: per (chunk, 1024-channel block) re-run the recurrence from the
// correct carry-in. x streams through a double-buffered LDS pipeline fed by
// CDNA5 async global->LDS b128 loads (ASYNCcnt-tracked). Output streams are
// written non-temporally so they don't evict x from L2.
// ---------------------------------------------------------------------------
__global__ void __launch_bounds__(256) lru_phase3(
    const float* __restrict__ x, const float* __restrict__ logd,
    const float* __restrict__ bias, const f32x4* __restrict__ carry,
    float* __restrict__ out, float* __restrict__ hout) {
    __shared__ __align__(16) float smem[2 * TT * CHB];   // 64 KB

    const int tid    = threadIdx.x;
    const int chBase = blockIdx.x * CHB;
    const int chunk  = blockIdx.y;
    const int ch0    = chBase + tid * 4;                 // first of 4 channels

    // chBase is a multiple of 1024 == D, so (ch0 % D)/4 == tid
    const f32x4 ld = ((const f32x4*)logd)[tid & (D / 4 - 1)];
    const f32x4 bb = ((const f32x4*)bias)[tid & (D / 4 - 1)];
    f32x4 dv;
#pragma unroll
    for (int j = 0; j < 4; ++j) dv[j] = sigf(ld[j]);

    f32x4 h = carry[(size_t)chunk * G4 + (size_t)(ch0 >> 2)];

    // low 32 bits of the flat LDS address == byte offset in the WG's LDS
    const uint32_t ldsBase = (uint32_t)(uintptr_t)(void*)smem;
    const int t0 = chunk * L;

    // one tile = TT rows x CHB channels = 32 KB = 2048 x 16B segments
    // -> 8 b128 async loads per thread per tile
    auto issue_tile = [&](int buf, int tile) {
        const char* gbase = (const char*)x +
            ((size_t)(t0 + tile * TT) * BD + (size_t)chBase) * sizeof(float);
        const uint32_t lbase = ldsBase + (uint32_t)buf * (TT * CHB * 4);
#pragma unroll
        for (int j = 0; j < 8; ++j) {
            const int s   = tid + j * 256;       // segment 0..2047
            const int tl  = s >> 8;              // row (256 segs per 4 KB row)
            const int col = (s & 255) << 4;      // byte column within row
            const char* gaddr = gbase + (size_t)tl * (BD * 4) + col;
            const uint32_t laddr = lbase + (uint32_t)(tl * (CHB * 4) + col);
            asm volatile("global_load_async_to_lds_b128 %0, %1, off"
                         :: "v"(laddr), "v"(gaddr) : "memory");
        }
    };

    issue_tile(0, 0);
    for (int tile = 0; tile < NTILE; ++tile) {
        if (tile + 1 < NTILE) {
            issue_tile((tile + 1) & 1, tile + 1);
            // 8 newer async ops outstanding; in-order retirement means
            // ASYNCcnt<=8 implies the current tile has landed in LDS
            asm volatile("s_wait_asynccnt 0x8" ::: "memory");
        } else {
            asm volatile("s_wait_asynccnt 0x0" ::: "memory");
        }
        __syncthreads();   // all waves' async data visible before LDS reads

        const f32x4* buf = (const f32x4*)&smem[(tile & 1) * (TT * CHB)];
        const int tb = t0 + tile * TT;
#pragma unroll
        for (int tl = 0; tl < TT; ++tl) {
            f32x4 xv = buf[tl * (CHB / 4) + tid];           // ds_load_b128
            h = dv * (xv + h) + bb;
            f32x4 sg;
#pragma unroll
            for (int j = 0; j < 4; ++j) sg[j] = fast_sig(h[j]);
            f32x4 o = h * h * sg;                            // h * silu(h)
            __builtin_nontemporal_store(
                o, (f32x4*)(out + (size_t)(tb + tl) * BD + ch0));
            __builtin_nontemporal_store(
                h, (f32x4*)(hout + (size_t)(tb + tl + 1) * BD + ch0));
        }
        __syncthreads();   // everyone done reading this buffer before reuse
    }
}

// ---------------------------------------------------------------------------
extern "C" void kernel_launch(void* const* d_in, const int* in_sizes, int n_in,
                              void* d_out, int out_size, void* d_ws, size_t ws_size,
                              hipStream_t stream) {
    (void)in_sizes; (void)n_in; (void)out_size; (void)ws_size;

    const float* x    = (const float*)d_in[0];   // [T,B,D]
    const float* h0   = (const float*)d_in[1];   // [B,D]
    const float* logd = (const float*)d_in[2];   // [D]
    const float* bias = (const float*)d_in[3];   // [D]

    float* out  = (float*)d_out;                 // [T,B,D]
    float* hout = out + (size_t)T * BD;          // [T+1,B,D]

    // workspace: sLast (1 MiB) + carry (1 MiB)
    f32x4* sLast = (f32x4*)d_ws;
    f32x4* carry = sLast + (size_t)CCH * G4;

    lru_phase1<<<dim3(G4 / 256, CCH), 256, 0, stream>>>(x, logd, bias, sLast);
    lru_phase2<<<dim3(G4 / 256), 256, 0, stream>>>(h0, logd, sLast, carry, hout);
    lru_phase3<<<dim3(BD / CHB, CCH), 256, 0, stream>>>(
        x, logd, bias, carry, out, hout);
}